// VectorQuantizationLayer_41772851921059
// MI455X (gfx1250) — compile-verified
//
#include <hip/hip_runtime.h>
#include <hip/hip_bf16.h>
#include <stdint.h>

typedef __attribute__((ext_vector_type(2))) float v2f;
typedef __attribute__((ext_vector_type(4))) float f4;
typedef __attribute__((ext_vector_type(8))) float v8f;

#define B_  8192
#define G_  4
#define K_  512
#define D_  512
#define LDS_PITCH 132   // 128 + 4 pad floats -> conflict-free b64 LDS reads

// ---------------------------------------------------------------------------
// Kernel 1: row squared-norms for latents (lat_sq[g*B+b]) and protos
// (proto_sq[g*K+k]). One wave per row, float4 coalesced loads.
// ---------------------------------------------------------------------------
__global__ __launch_bounds__(256) void k_rowsq(const float* __restrict__ lat,
                                               const float* __restrict__ proto,
                                               float* __restrict__ lat_sq,
                                               float* __restrict__ proto_sq) {
    const int wid  = threadIdx.x >> 5;
    const int lane = threadIdx.x & 31;
    const int r = blockIdx.x * 8 + wid;

    const float* src;
    float* dst;
    if (r < G_ * B_) {
        const int g = r / B_;
        const int b = r % B_;
        src = lat + ((size_t)b * G_ + g) * D_;
        dst = lat_sq + r;
    } else {
        const int r2 = r - G_ * B_;
        if (r2 >= G_ * K_) return;
        src = proto + (size_t)r2 * D_;
        dst = proto_sq + r2;
    }

    const f4* s4 = (const f4*)src;
    float s = 0.0f;
#pragma unroll
    for (int i = 0; i < 4; ++i) {           // 4 * 32 lanes * 4 floats = 512
        f4 v = s4[lane + 32 * i];
        s += v.x * v.x + v.y * v.y + v.z * v.z + v.w * v.w;
    }
#pragma unroll
    for (int m = 16; m >= 1; m >>= 1) s += __shfl_xor(s, m);
    if (lane == 0) *dst = s;
}

// ---------------------------------------------------------------------------
// Kernel 2: WMMA GEMM  cross = latents[:,g,:] @ proto[g]^T  fused with
// dists = lat_sq + proto_sq - 2*cross, plus per-row argmin over K.
// Block = 256 threads = 8 waves; block covers (g, 128 rows of B).
// Wave w -> 16-row strip; 4 accumulators (64 K-columns) per wave.
// Proto tiles double-buffered in LDS via CDNA5 async-to-LDS loads
// (GLOBAL_LOAD_ASYNC_TO_LDS_B128, ASYNCcnt), overlapped with the WMMA burst.
// ---------------------------------------------------------------------------
__global__ __launch_bounds__(256) void k_main(const float* __restrict__ lat,
                                              const float* __restrict__ proto,
                                              const float* __restrict__ lat_sq,
                                              const float* __restrict__ proto_sq,
                                              float* __restrict__ dists,
                                              int* __restrict__ idx_ws) {
    __shared__ float ptile[2][64 * LDS_PITCH];  // 2 x (64 proto rows x 128 d)

    const int g    = blockIdx.x & (G_ - 1);
    const int rb   = (blockIdx.x >> 2) * 128;
    const int tid  = threadIdx.x;
    const int wid  = tid >> 5;
    const int lane = tid & 31;
    const int ln   = lane & 15;
    const int hi   = lane >> 4;

    // fp32 WMMA 16x16x4 A layout: lane<16 -> (M=lane, K=0..1),
    // lane>=16 -> (M=lane-16, K=2..3)  => float2 load at d + 2*hi.
    const int row_a = rb + wid * 16 + ln;
    const float* aptr = lat + ((size_t)row_a * G_ + g) * D_;
    const float* pg   = proto + (size_t)g * K_ * D_;

    // lat_sq for the 8 (M = j + 8*hi) rows this lane's C fragment covers
    float lsq[8];
#pragma unroll
    for (int j = 0; j < 8; ++j)
        lsq[j] = lat_sq[g * B_ + rb + wid * 16 + j + 8 * hi];

    float best_d[8];
    int   best_k[8];
#pragma unroll
    for (int j = 0; j < 8; ++j) { best_d[j] = 3.4e38f; best_k[j] = 0; }

    // Per-lane (prow, pc4) slots for cooperative staging: 2048 float4s / 256 thr
    // Issue CDNA5 async loads straight into LDS (no VGPR round-trip).
    auto stage = [&](int buf, int kbase, int dbase) {
#pragma unroll
        for (int i = 0; i < 8; ++i) {
            const int linear = tid + 256 * i;       // 0..2047
            const int prow   = linear >> 5;
            const int pc4    = linear & 31;
            // low 32 bits of the flat shared address == LDS byte offset
            unsigned ldsoff = (unsigned)(uintptr_t)&ptile[buf][prow * LDS_PITCH + pc4 * 4];
            const float* ga = pg + (size_t)(kbase + prow) * D_ + dbase + pc4 * 4;
            asm volatile("global_load_async_to_lds_b128 %0, %1, off"
                         :: "v"(ldsoff), "v"(ga) : "memory");
        }
    };

    for (int kg = 0; kg < 8; ++kg) {          // 8 groups of 64 K-columns
        const int kbase = kg * 64;

        v8f acc[4];
#pragma unroll
        for (int t = 0; t < 4; ++t)
#pragma unroll
            for (int e = 0; e < 8; ++e) acc[t][e] = 0.0f;

        // prologue: stage first d-block into buffer 0
        stage(0, kbase, 0);
        asm volatile("s_wait_asynccnt 0x0" ::: "memory");
        __syncthreads();

        for (int db = 0; db < 4; ++db) {      // 4 d-blocks of 128
            const int p = db & 1;
            if (db < 3) stage(1 - p, kbase, (db + 1) * 128);  // overlap w/ WMMAs

            const float* ap = aptr + db * 128 + 2 * hi;
#pragma unroll 8
            for (int c = 0; c < 32; ++c) {    // 32 K=4 chunks per d-block
                v2f a = *(const v2f*)(ap + c * 4);
#pragma unroll
                for (int t = 0; t < 4; ++t) {
                    v2f b = *(const v2f*)(&ptile[p][(t * 16 + ln) * LDS_PITCH + c * 4 + 2 * hi]);
                    acc[t] = __builtin_amdgcn_wmma_f32_16x16x4_f32(
                        /*neg_a=*/false, a, /*neg_b=*/false, b,
                        /*c_mod=*/(short)0, acc[t],
                        /*reuse_a=*/false, /*reuse_b=*/false);
                }
            }
            asm volatile("s_wait_asynccnt 0x0" ::: "memory");
            __syncthreads();
        }

        // epilogue: dists + running argmin for the 4 finished tiles
#pragma unroll
        for (int t = 0; t < 4; ++t) {
            const int col = kbase + t * 16 + ln;
            const float psq = proto_sq[g * K_ + col];
#pragma unroll
            for (int j = 0; j < 8; ++j) {
                const int row = rb + wid * 16 + j + 8 * hi;
                const float dv = lsq[j] + psq - 2.0f * acc[t][j];
                dists[((size_t)row * G_ + g) * K_ + col] = dv;
                if (dv < best_d[j]) { best_d[j] = dv; best_k[j] = col; }
            }
        }
    }

    // argmin reduction across the 16 lanes sharing each row (first-index ties)
#pragma unroll
    for (int j = 0; j < 8; ++j) {
        float bd = best_d[j];
        int   bk = best_k[j];
#pragma unroll
        for (int m = 1; m <= 8; m <<= 1) {
            float od = __shfl_xor(bd, m);
            int   ok = __shfl_xor(bk, m);
            if (od < bd || (od == bd && ok < bk)) { bd = od; bk = ok; }
        }
        if (ln == 0)
            idx_ws[g * B_ + rb + wid * 16 + j + 8 * hi] = bk;
    }
}

// ---------------------------------------------------------------------------
// Kernel 3: gather quant = proto[g, idx], write quantized_recon (== quant),
// and per-(b,g) partial sum of (quant - lat)^2.
// ---------------------------------------------------------------------------
__global__ __launch_bounds__(128) void k_gather(const float* __restrict__ lat,
                                                const float* __restrict__ proto,
                                                const int* __restrict__ idx_ws,
                                                float* __restrict__ recon,
                                                float* __restrict__ partial) {
    __shared__ float red[128];
    const int b = blockIdx.x >> 2;
    const int g = blockIdx.x & 3;
    const int t = threadIdx.x;

    const int k = idx_ws[g * B_ + b];
    const f4* q4 = (const f4*)(proto + ((size_t)g * K_ + k) * D_);
    const f4* l4 = (const f4*)(lat + ((size_t)b * G_ + g) * D_);
    f4* r4 = (f4*)(recon + ((size_t)b * G_ + g) * D_);

    f4 q = q4[t];
    f4 l = l4[t];
    r4[t] = q;
    f4 d = q - l;
    red[t] = d.x * d.x + d.y * d.y + d.z * d.z + d.w * d.w;
    __syncthreads();
#pragma unroll
    for (int off = 64; off >= 1; off >>= 1) {
        if (t < off) red[t] += red[t + off];
        __syncthreads();
    }
    if (t == 0) partial[blockIdx.x] = red[0];
}

// ---------------------------------------------------------------------------
// Kernel 4: deterministic fixed-order reduction of partials -> vq_loss.
// commitment == embedding numerically, so
// vq_loss = total * (1 + BETA) / (B * D * G) = total * 1.25 / 16777216.
// ---------------------------------------------------------------------------
__global__ __launch_bounds__(256) void k_loss(const float* __restrict__ partial,
                                              float* __restrict__ vq_out) {
    __shared__ float red[256];
    const int t = threadIdx.x;
    float s = 0.0f;
    for (int i = t; i < B_ * G_; i += 256) s += partial[i];
    red[t] = s;
    __syncthreads();
#pragma unroll
    for (int off = 128; off >= 1; off >>= 1) {
        if (t < off) red[t] += red[t + off];
        __syncthreads();
    }
    if (t == 0) *vq_out = red[0] * (1.25f / 16777216.0f);
}

extern "C" void kernel_launch(void* const* d_in, const int* in_sizes, int n_in,
                              void* d_out, int out_size, void* d_ws, size_t ws_size,
                              hipStream_t stream) {
    const float* lat   = (const float*)d_in[0];   // (B, G, D) f32
    const float* proto = (const float*)d_in[1];   // (G, K, D) f32

    float* out   = (float*)d_out;
    float* recon = out;                               // B*G*D
    float* vq    = out + (size_t)B_ * G_ * D_;        // 1
    float* dists = vq + 1;                            // B*G*K

    float* ws       = (float*)d_ws;
    float* lat_sq   = ws;                 // G*B   = 32768 floats
    float* proto_sq = ws + 32768;         // G*K   =  2048 floats
    float* partial  = ws + 34816;         // B*G   = 32768 floats
    int*   idx_ws   = (int*)(ws + 67584); // B*G   = 32768 ints

    k_rowsq <<<(G_ * B_ + G_ * K_) / 8, 256, 0, stream>>>(lat, proto, lat_sq, proto_sq);
    k_main  <<<(B_ / 128) * G_,       256, 0, stream>>>(lat, proto, lat_sq, proto_sq, dists, idx_ws);
    k_gather<<<B_ * G_,               128, 0, stream>>>(lat, proto, idx_ws, recon, partial);
    k_loss  <<<1,                     256, 0, stream>>>(partial, vq);
}